// HAMConvBlock_35330400977382
// MI455X (gfx1250) — compile-verified
//
#include <hip/hip_runtime.h>
#include <hip/hip_bf16.h>
#include <math.h>

// ---------------- problem constants ----------------
constexpr int B_  = 16;
constexpr int I_  = 64;
constexpr int C_  = 128;
constexpr int L_  = 16384;
constexpr int KS_ = 5;
constexpr int KTOT = I_ * KS_;     // 320, flattened K (i fastest within a tap)
constexpr int KCR  = 76;
constexpr float EPS_ = 1e-5f;

// ---------------- workspace layout (bytes) ----------------
constexpr size_t OFF_WR    = 0;                       // 128*320 f32 = 163840
constexpr size_t OFF_SCALE = 163840;                  // 128 f32
constexpr size_t OFF_SHIFT = 164352;                  // 128 f32
constexpr size_t OFF_GSUM  = 164864;                  // 16*128 f32
constexpr size_t OFF_GMAX  = 173056;                  // 16*128 u32 (float bits, vals >= 0)
constexpr size_t OFF_CMAP  = 181248;                  // 16*128 f32
constexpr size_t OFF_CMASK = 189440;                  // 16*128 f32
constexpr size_t OFF_F     = 204800;                  // B*C*L f32 = 134217728
// total ~134.4 MB required in d_ws

typedef float v2f __attribute__((ext_vector_type(2)));
typedef float v8f __attribute__((ext_vector_type(8)));

// ---------------- kernel 0: setup ----------------
// Wr[c][k*64+i] = conv_w[c][i][k]; fold conv bias + BN into scale/shift; zero accumulators.
__global__ void ham_setup_kernel(const float* __restrict__ conv_w,
                                 const float* __restrict__ conv_b,
                                 const float* __restrict__ bn_g,
                                 const float* __restrict__ bn_b,
                                 const float* __restrict__ bn_m,
                                 const float* __restrict__ bn_v,
                                 float* __restrict__ Wr,
                                 float* __restrict__ scale,
                                 float* __restrict__ shift,
                                 float* __restrict__ gsum,
                                 unsigned* __restrict__ gmax) {
    int t = blockIdx.x * 256 + threadIdx.x;
    if (t < C_ * KTOT) {
        int c = t / KTOT;
        int r = t - c * KTOT;          // r = k*64 + i
        int k = r >> 6;
        int i = r & 63;
        Wr[t] = conv_w[c * (I_ * KS_) + i * KS_ + k];
    }
    if (t < B_ * C_) {
        gsum[t] = 0.0f;
        gmax[t] = 0u;                  // bits of +0.0f; post-relu values are >= 0
    }
    if (t < C_) {
        float inv = bn_g[t] / sqrtf(bn_v[t] + EPS_);
        scale[t] = inv;
        shift[t] = (conv_b[t] - bn_m[t]) * inv + bn_b[t];
    }
}

// ---------------- kernel 1: conv + bn + relu + global sum/max, via f32 WMMA ----------------
// grid: B * (L/128) blocks, 256 threads (8 waves). Wave w owns channels [16w, 16w+16).
// Each wave computes 8 independent 16x16 (C x L) tiles, K-loop of 80 chunks of 4.
// X tile is stored COLUMN-major in LDS (Xs[col][i], pitch 68) so each B fragment
// (two consecutive input channels at one column) is a single conflict-free ds_load_b64.
constexpr int LT_    = 128;            // l-extent per block
constexpr int XCOLS  = LT_ + 4;        // 132 (halo 2 each side)
constexpr int XP2    = 68;             // floats per column: 64 i + 4 pad (bank spread)

__global__ void __launch_bounds__(256)
ham_conv_wmma_kernel(const float* __restrict__ x,
                     const float* __restrict__ Wr,
                     const float* __restrict__ scale,
                     const float* __restrict__ shift,
                     float* __restrict__ f,
                     float* __restrict__ gsum,
                     unsigned* __restrict__ gmax) {
    __shared__ float Xs[XCOLS * XP2];  // 132*68*4 = 35904 B

    const int b    = blockIdx.x >> 7;      // 128 l-blocks per batch
    const int lblk = blockIdx.x & 127;
    const int l0   = lblk * LT_;
    const int tid  = threadIdx.x;

    // cooperative X tile load (with zero halo); global reads coalesced along col
    const float* xb = x + (size_t)b * I_ * L_;
    for (int t = tid; t < I_ * XCOLS; t += 256) {
        int i   = t / XCOLS;
        int col = t - i * XCOLS;
        int l   = l0 + col - 2;
        Xs[col * XP2 + i] = (l >= 0 && l < L_) ? xb[i * L_ + l] : 0.0f;
    }
    __syncthreads();

    const int lane = tid & 31;
    const int wave = tid >> 5;
    const int half = lane >> 4;            // 0: lanes 0-15, 1: lanes 16-31
    const int ln   = lane & 15;
    const int c0   = wave * 16;

    v8f acc[8];
    #pragma unroll
    for (int n0 = 0; n0 < 8; ++n0) {
        v8f z = {0.f, 0.f, 0.f, 0.f, 0.f, 0.f, 0.f, 0.f};
        acc[n0] = z;
    }

    // A fragment: lane<16 holds W[m][kb+0..1], lane>=16 holds W[m][kb+2..3]  (ISA 16x4 A layout)
    const float* wrow = Wr + (size_t)(c0 + ln) * KTOT + 2 * half;
    const int rofs = 2 * half;             // B fragment: rows (K0,K1) low half / (K2,K3) high half

    for (int kb = 0; kb < KTOT; kb += 4) {
        const int k  = kb >> 6;            // conv tap 0..4
        const int ib = kb & 63;            // input-channel base of this chunk
        v2f a = *reinterpret_cast<const v2f*>(wrow + kb);       // global_load_b64
        const float* xcol = Xs + ib + rofs;
        #pragma unroll
        for (int n0 = 0; n0 < 8; ++n0) {
            const int col = n0 * 16 + ln + k;
            v2f bb = *reinterpret_cast<const v2f*>(xcol + col * XP2);  // ds_load_b64
            acc[n0] = __builtin_amdgcn_wmma_f32_16x16x4_f32(
                false, a, false, bb, (short)0, acc[n0], false, false);
        }
    }

    // epilogue: bn + relu, store f, accumulate per-channel sum/max
    float scl[8], sht[8], rsum[8], rmax[8];
    #pragma unroll
    for (int j = 0; j < 8; ++j) {
        int c = c0 + j + half * 8;         // C/D layout: VGPR j = rows j (lanes0-15) / j+8
        scl[j] = scale[c];
        sht[j] = shift[c];
        rsum[j] = 0.0f;
        rmax[j] = 0.0f;
    }

    float* fb = f + (size_t)b * C_ * L_;
    #pragma unroll
    for (int n0 = 0; n0 < 8; ++n0) {
        #pragma unroll
        for (int j = 0; j < 8; ++j) {
            float v = acc[n0][j] * scl[j] + sht[j];
            v = fmaxf(v, 0.0f);
            fb[(size_t)(c0 + j + half * 8) * L_ + (l0 + n0 * 16 + ln)] = v;
            rsum[j] += v;
            rmax[j] = fmaxf(rmax[j], v);
        }
    }

    // 16-lane tree reduction (stays within each half of the wave32)
    #pragma unroll
    for (int j = 0; j < 8; ++j) {
        #pragma unroll
        for (int m = 1; m < 16; m <<= 1) {
            rsum[j] += __shfl_xor(rsum[j], m, 32);
            rmax[j] = fmaxf(rmax[j], __shfl_xor(rmax[j], m, 32));
        }
    }
    if (ln == 0) {
        #pragma unroll
        for (int j = 0; j < 8; ++j) {
            int c = c0 + j + half * 8;
            atomicAdd(&gsum[b * C_ + c], rsum[j]);
            atomicMax(&gmax[b * C_ + c], __float_as_uint(rmax[j]));  // valid: vals >= 0
        }
    }
}

// ---------------- kernel 2: channel attention + top-k mask ----------------
// grid = B blocks, 128 threads.
__global__ void ham_attn_topk_kernel(const float* __restrict__ gsum,
                                     const unsigned* __restrict__ gmax,
                                     const float* __restrict__ alpha_p,
                                     const float* __restrict__ beta_p,
                                     const float* __restrict__ ca_w,
                                     const float* __restrict__ ca_b,
                                     float* __restrict__ cmap,
                                     float* __restrict__ cmask) {
    __shared__ float fa[C_];
    __shared__ float cm[C_];
    const int b = blockIdx.x;
    const int c = threadIdx.x;

    float avg = gsum[b * C_ + c] * (1.0f / (float)L_);
    float mx  = __uint_as_float(gmax[b * C_ + c]);
    float al  = alpha_p[0], be = beta_p[0];
    // alpha*avg + 0.5*(avg+max) + beta*max
    fa[c] = (al + 0.5f) * avg + (be + 0.5f) * mx;
    __syncthreads();

    float z = ca_b[0];
    #pragma unroll
    for (int t = 0; t < 5; ++t) {
        int j = c + t - 2;
        if (j >= 0 && j < C_) z += ca_w[t] * fa[j];
    }
    float s = 1.0f / (1.0f + expf(-z));
    cm[c] = s;
    cmap[b * C_ + c] = s;
    __syncthreads();

    // rank with jax top_k tie-breaking (lower index wins ties)
    int cnt = 0;
    for (int j = 0; j < C_; ++j) {
        float vj = cm[j];
        cnt += (vj > s) || (vj == s && j < c);
    }
    cmask[b * C_ + c] = (cnt < KCR) ? 1.0f : 0.0f;
}

// ---------------- kernel 3: fused crf, channel reductions, gate conv, output ----------------
// grid = B * (L/64) blocks, 256 threads. Tile: 64 l + 3-halo each side.
constexpr int TL_   = 64;
constexpr int COLS_ = TL_ + 6;         // 70

__global__ void __launch_bounds__(256)
ham_fuse_out_kernel(const float* __restrict__ f,
                    const float* __restrict__ cmap,
                    const float* __restrict__ cmask,
                    const float* __restrict__ sa_w,
                    const float* __restrict__ sabn_g,
                    const float* __restrict__ sabn_b,
                    const float* __restrict__ sabn_m,
                    const float* __restrict__ sabn_v,
                    float* __restrict__ out) {
    __shared__ float Fs[C_][COLS_];    // crf = f * channel_map
    __shared__ float cmS[C_], mkS[C_];
    __shared__ float mxA[COLS_], avA[COLS_], smxA[COLS_], savA[COLS_];
    __shared__ float A1s[TL_], A2s[TL_];

    const int b    = blockIdx.x >> 8;  // 256 l-tiles per batch
    const int lblk = blockIdx.x & 255;
    const int l0   = lblk * TL_;
    const int tid  = threadIdx.x;

    if (tid < C_) {
        cmS[tid] = cmap[b * C_ + tid];
        mkS[tid] = cmask[b * C_ + tid];
    }
    __syncthreads();

    const float* fb = f + (size_t)b * C_ * L_;
    for (int t = tid; t < C_ * COLS_; t += 256) {
        int c   = t / COLS_;
        int col = t - c * COLS_;
        int l   = l0 + col - 3;
        float v = 0.0f;
        if (l >= 0 && l < L_) v = fb[(size_t)c * L_ + l] * cmS[c];
        Fs[c][col] = v;                // zero outside == conv zero padding
    }
    __syncthreads();

    if (tid < COLS_) {
        float cmx = 0.0f, cs = 0.0f, smx = 0.0f, ss = 0.0f;
        for (int c = 0; c < C_; ++c) {
            float v = Fs[c][tid];
            if (mkS[c] > 0.5f) { cmx = fmaxf(cmx, v); cs += v; }
            else               { smx = fmaxf(smx, v); ss += v; }
        }
        mxA[tid]  = cmx;                       // max over cf (zeros elsewhere, v>=0)
        avA[tid]  = cs * (1.0f / (float)KCR);  // mean*(C/KCR) == sum/KCR
        smxA[tid] = smx;
        savA[tid] = ss * (1.0f / (float)(C_ - KCR));
    }
    __syncthreads();

    if (tid < TL_) {
        float inv = sabn_g[0] / sqrtf(sabn_v[0] + EPS_);
        float sb  = sabn_b[0] - sabn_m[0] * inv;
        float y1 = 0.0f, y2 = 0.0f;
        #pragma unroll
        for (int t = 0; t < 7; ++t) {
            int col = tid + t;                 // (tid+3) + (t-3)
            y1 += sa_w[t] * mxA[col]  + sa_w[7 + t] * avA[col];
            y2 += sa_w[t] * smxA[col] + sa_w[7 + t] * savA[col];
        }
        y1 = fmaxf(y1 * inv + sb, 0.0f);
        y2 = fmaxf(y2 * inv + sb, 0.0f);
        A1s[tid] = 1.0f / (1.0f + expf(-y1));
        A2s[tid] = 1.0f / (1.0f + expf(-y2));
    }
    __syncthreads();

    float* ob = out + (size_t)b * C_ * L_ + l0;
    #pragma unroll 4
    for (int it = 0; it < (C_ * TL_) / 256; ++it) {
        int idx = it * 256 + tid;
        int c   = idx >> 6;
        int li  = idx & 63;
        float crf = Fs[c][li + 3];
        float a   = (mkS[c] > 0.5f) ? A1s[li] : A2s[li];
        ob[(size_t)c * L_ + li] = crf * a;
    }
}

// ---------------- launcher ----------------
extern "C" void kernel_launch(void* const* d_in, const int* in_sizes, int n_in,
                              void* d_out, int out_size, void* d_ws, size_t ws_size,
                              hipStream_t stream) {
    const float* x       = (const float*)d_in[0];
    const float* conv_w  = (const float*)d_in[1];
    const float* conv_b  = (const float*)d_in[2];
    const float* bn_g    = (const float*)d_in[3];
    const float* bn_b    = (const float*)d_in[4];
    const float* bn_m    = (const float*)d_in[5];
    const float* bn_v    = (const float*)d_in[6];
    const float* alpha   = (const float*)d_in[7];
    const float* beta    = (const float*)d_in[8];
    const float* ca_w    = (const float*)d_in[9];
    const float* ca_b    = (const float*)d_in[10];
    const float* sa_w    = (const float*)d_in[11];
    const float* sabn_g  = (const float*)d_in[12];
    const float* sabn_b  = (const float*)d_in[13];
    const float* sabn_m  = (const float*)d_in[14];
    const float* sabn_v  = (const float*)d_in[15];
    float* out = (float*)d_out;

    char* ws = (char*)d_ws;
    float*    Wr    = (float*)(ws + OFF_WR);
    float*    scale = (float*)(ws + OFF_SCALE);
    float*    shift = (float*)(ws + OFF_SHIFT);
    float*    gsum  = (float*)(ws + OFF_GSUM);
    unsigned* gmax  = (unsigned*)(ws + OFF_GMAX);
    float*    cmap  = (float*)(ws + OFF_CMAP);
    float*    cmask = (float*)(ws + OFF_CMASK);
    float*    fbuf  = (float*)(ws + OFF_F);     // B*C*L f32 (~134 MB)

    // 0: setup (also zeroes accumulators every call — required for graph replay)
    ham_setup_kernel<<<(C_ * KTOT + 255) / 256, 256, 0, stream>>>(
        conv_w, conv_b, bn_g, bn_b, bn_m, bn_v, Wr, scale, shift, gsum, gmax);

    // 1: conv + bn + relu + per-(b,c) sum/max, f32 WMMA
    ham_conv_wmma_kernel<<<B_ * (L_ / LT_), 256, 0, stream>>>(
        x, Wr, scale, shift, fbuf, gsum, gmax);

    // 2: channel attention + top-76 mask
    ham_attn_topk_kernel<<<B_, C_, 0, stream>>>(
        gsum, gmax, alpha, beta, ca_w, ca_b, cmap, cmask);

    // 3: fused reductions + gate + output
    ham_fuse_out_kernel<<<B_ * (L_ / TL_), 256, 0, stream>>>(
        fbuf, cmap, cmask, sa_w, sabn_g, sabn_b, sabn_m, sabn_v, out);
}